// Extractor_39032662786373
// MI455X (gfx1250) — compile-verified
//
#include <hip/hip_runtime.h>
#include <hip/hip_bf16.h>

#define IN_DIM 10
#define KPAD   12
#define HIDDEN 16

typedef __attribute__((ext_vector_type(2))) float v2f;
typedef __attribute__((ext_vector_type(8))) float v8f;

// ---------------------------------------------------------------------------
// Kernel 1: zero the accumulator workspace (summed [N*IN] ++ counts [N])
// ---------------------------------------------------------------------------
__global__ void zero_ws_kernel(float* __restrict__ p, int n) {
    int i = blockIdx.x * blockDim.x + threadIdx.x;
    if (i < n) p[i] = 0.0f;
}

// ---------------------------------------------------------------------------
// Kernel 2: edge aggregation. One thread per edge; 51MB of index stream from
// HBM, gather/scatter into the L2-resident x (4MB) and accumulator (4.4MB).
// Bound by L2 f32-atomic throughput -> plain global_atomic_add_f32, no return.
// ---------------------------------------------------------------------------
__global__ void aggregate_kernel(const float* __restrict__ x,
                                 const int* __restrict__ edge_index,
                                 float* __restrict__ summed,
                                 float* __restrict__ counts,
                                 int n_edges) {
    int e = blockIdx.x * blockDim.x + threadIdx.x;
    if (e >= n_edges) return;
    int src = edge_index[e];            // edge_index[0, e]
    int dst = edge_index[n_edges + e];  // edge_index[1, e]
    const float* xs = x + (size_t)src * IN_DIM;
    float* sd = summed + (size_t)dst * IN_DIM;
#pragma unroll
    for (int k = 0; k < IN_DIM; k += 2) {
        float2 v = *(const float2*)(xs + k);   // rows are 40B -> 8B aligned
        atomicAdd(sd + k,     v.x);
        atomicAdd(sd + k + 1, v.y);
    }
    atomicAdd(counts + dst, 1.0f);
}

// ---------------------------------------------------------------------------
// Kernel 3: WMMA epilogue. One wave per 16-node tile:
//   D = mean[16x10] @ W_l[10x16] + x[16x10] @ W_r[10x16] + b_l
// Data path: coalesced global_load -> zero-padded LDS tiles -> ds_load_2addr
// pairs -> 6 chained V_WMMA_F32_16X16X4_F32 -> unguarded coalesced stores
// (boundary handled by a wave-uniform branch, not per-lane EXEC masking).
//
// WMMA operand layouts (ISA 7.12.2):
//   A 16x4 f32 : M = lane&15; K-pair = {0,1} (lanes 0-15) / {2,3} (lanes 16-31)
//   B 4x16 f32 : N = lane&15; same K-pair selection
//   C/D 16x16  : N = lane&15; VGPR r -> M = r + (lane>=16 ? 8 : 0)
// ---------------------------------------------------------------------------
__global__ void sage_wmma_kernel(const float* __restrict__ x,
                                 const float* __restrict__ summed,
                                 const float* __restrict__ counts,
                                 const float* __restrict__ W_l,
                                 const float* __restrict__ b_l,
                                 const float* __restrict__ W_r,
                                 float* __restrict__ out,
                                 int n_nodes) {
    // Padded weights: sW[wsel][k][n], k in [0,12), rows 10..11 zeroed.
    __shared__ float sW[2 * KPAD * HIDDEN];           // 1536 B
    // Per-wave A tiles: [wave][0=mean-src,1=x][row][KPAD], rows 16, cols 12.
    __shared__ float sA[4 * 2 * 16 * KPAD];           // 6144 B

    const int lane = threadIdx.x & 31;
    const int wave = threadIdx.x >> 5;

    // ---- stage padded weights (all waves participate, then barrier) ----
    for (int idx = threadIdx.x; idx < 2 * KPAD * HIDDEN; idx += blockDim.x) {
        const int wsel = idx / (KPAD * HIDDEN);
        const int rem  = idx - wsel * (KPAD * HIDDEN);
        const int k    = rem >> 4;
        const int n    = rem & 15;
        const float* W = wsel ? W_r : W_l;
        sW[idx] = (k < IN_DIM) ? W[k * HIDDEN + n] : 0.0f;
    }
    __syncthreads();

    const int tile     = blockIdx.x * (blockDim.x >> 5) + wave;
    const int row_base = tile * 16;
    if (row_base >= n_nodes) return;   // uniform per wave; after the barrier

    const int m     = lane & 15;        // A row in tile == B/C/D column
    const int khalf = (lane >> 4) * 2;  // 0 (lanes 0-15) or 2 (lanes 16-31)

    // ---- stage A tiles: 16 contiguous rows = 640B, 5 coalesced b32/lane ----
    float* myA = &sA[wave * (2 * 16 * KPAD)];
    const int n_total = n_nodes * IN_DIM;
    const int base    = row_base * IN_DIM;
#pragma unroll
    for (int j = 0; j < 5; ++j) {
        const int  f  = lane + 32 * j;              // 0..159 flat in tile
        const int  g  = base + f;
        const int  gc = (g < n_total) ? g : (n_total - 1);  // clamped addr
        float vs = summed[gc];
        float vx = x[gc];
        if (g >= n_total) { vs = 0.0f; vx = 0.0f; } // v_cndmask, no branch
        const int lr = f / 10;
        const int lc = f - lr * 10;
        const int o  = lr * KPAD + lc;
        myA[o]             = vs;
        myA[16 * KPAD + o] = vx;
    }
    // zero the K=10,11 pad columns: lane -> (tile t = lane>>4, row = lane&15)
    {
        float* pz = &myA[(lane >> 4) * (16 * KPAD) + (lane & 15) * KPAD + IN_DIM];
        *(float2*)pz = make_float2(0.0f, 0.0f);     // ds_store_b64
    }

    // ---- per-row 1/deg for the mean (A rows beyond N are already zero) ----
    const int  rowm = row_base + m;
    const int  rc   = (rowm < n_nodes) ? rowm : (n_nodes - 1);
    const float cnt = counts[rc];
    const float inv = 1.0f / fmaxf(cnt, 1.0f);

    // ---- C init with bias (depends only on column N == m) ----
    const float bias = b_l[m];
    v8f acc;
#pragma unroll
    for (int i = 0; i < 8; ++i) acc[i] = bias;

    // ---- mean @ W_l : 3 K-steps ----
#pragma unroll
    for (int s = 0; s < 3; ++s) {
        const int k0 = khalf + 4 * s;
        const float2 av = *(const float2*)&myA[m * KPAD + k0];  // ds_load_b64
        v2f a; a.x = av.x * inv; a.y = av.y * inv;
        v2f b; b.x = sW[k0 * HIDDEN + m]; b.y = sW[(k0 + 1) * HIDDEN + m];
        acc = __builtin_amdgcn_wmma_f32_16x16x4_f32(false, a, false, b,
                                                    (short)0, acc, false, false);
    }

    // ---- x @ W_r : 3 K-steps ----
#pragma unroll
    for (int s = 0; s < 3; ++s) {
        const int k0 = khalf + 4 * s;
        const float2 av = *(const float2*)&myA[16 * KPAD + m * KPAD + k0];
        v2f a; a.x = av.x; a.y = av.y;
        v2f b; b.x = sW[KPAD * HIDDEN + k0 * HIDDEN + m];
        b.y = sW[KPAD * HIDDEN + (k0 + 1) * HIDDEN + m];
        acc = __builtin_amdgcn_wmma_f32_16x16x4_f32(false, a, false, b,
                                                    (short)0, acc, false, false);
    }

    // ---- store D: VGPR rv -> row rv + (lane>=16 ? 8 : 0), col m ----
    // Boundary is wave-uniform: full tiles take the unguarded path with one
    // base address + immediate offsets (8x global_store_b32, coalesced).
    float* po = out + (size_t)(row_base + ((lane >> 4) << 3)) * HIDDEN + m;
    if (row_base + 16 <= n_nodes) {
#pragma unroll
        for (int rv = 0; rv < 8; ++rv)
            po[rv * HIDDEN] = acc[rv];
    } else {
#pragma unroll
        for (int rv = 0; rv < 8; ++rv) {
            const int out_m = rv + ((lane >> 4) << 3);
            if (row_base + out_m < n_nodes) po[rv * HIDDEN] = acc[rv];
        }
    }
}

// ---------------------------------------------------------------------------
extern "C" void kernel_launch(void* const* d_in, const int* in_sizes, int n_in,
                              void* d_out, int out_size, void* d_ws, size_t ws_size,
                              hipStream_t stream) {
    const float* x   = (const float*)d_in[0];
    const int*   ei  = (const int*)d_in[1];
    const float* W_l = (const float*)d_in[2];
    const float* b_l = (const float*)d_in[3];
    const float* W_r = (const float*)d_in[4];
    float*       out = (float*)d_out;

    const int n_nodes = in_sizes[0] / IN_DIM;
    const int n_edges = in_sizes[1] / 2;

    float* summed = (float*)d_ws;                        // [n_nodes * IN_DIM]
    float* counts = summed + (size_t)n_nodes * IN_DIM;   // [n_nodes]

    // 1) zero accumulators (sum + counts contiguous: 11 floats per node)
    const int zn = n_nodes * (IN_DIM + 1);
    zero_ws_kernel<<<(zn + 255) / 256, 256, 0, stream>>>(summed, zn);

    // 2) edge scatter-add (bottleneck: HBM index stream + L2 atomics)
    aggregate_kernel<<<(n_edges + 255) / 256, 256, 0, stream>>>(
        x, ei, summed, counts, n_edges);

    // 3) WMMA epilogue: one wave per 16-node tile, 4 waves / 128-thread block
    const int tiles = (n_nodes + 15) / 16;
    const int waves_per_block = 4;
    const int blocks = (tiles + waves_per_block - 1) / waves_per_block;
    sage_wmma_kernel<<<blocks, waves_per_block * 32, 0, stream>>>(
        x, summed, counts, W_l, b_l, W_r, out, n_nodes);
}